// EdgeLayer_1142461300898
// MI455X (gfx1250) — compile-verified
//
#include <hip/hip_runtime.h>
#include <hip/hip_bf16.h>
#include <math.h>

#define H_DIM 96
#define BN_EPS 1e-5f
#define WLDS_STRIDE 104   // pad: 2*104 mod 64 = 16 -> half-wave B rows hit disjoint LDS banks

typedef float v2f __attribute__((ext_vector_type(2)));
typedef float v8f __attribute__((ext_vector_type(8)));

// ---- monotonic float <-> uint encoding so we can use native integer atomicMax ----
__device__ __forceinline__ unsigned f2ord(float f) {
    unsigned u = __float_as_uint(f);
    return (u & 0x80000000u) ? ~u : (u | 0x80000000u);
}
__device__ __forceinline__ float ord2f(unsigned u) {
    return (u & 0x80000000u) ? __uint_as_float(u & 0x7FFFFFFFu)
                             : __uint_as_float(~u);
}

// K0: zero-init contiguous scratch region (seg_max, seg_sum, neigh, bn accumulators)
__global__ void k0_zero(unsigned* __restrict__ p, long long n) {
    long long i = (long long)blockIdx.x * blockDim.x + threadIdx.x;
    long long stride = (long long)gridDim.x * blockDim.x;
    for (; i < n; i += stride) p[i] = 0u;
}

// K1: wave-per-edge score = dot(rel_emb[rel_id[e]], ent_emb[dst[e]]);
//     atomicMax into per-destination segment max (ordered-uint encoding).
__global__ void k1_score(const float* __restrict__ ent, const float* __restrict__ rel,
                         const int* __restrict__ rel_id, const int* __restrict__ dst,
                         float* __restrict__ score, unsigned* __restrict__ seg_max_u,
                         int n_edge) {
    int wave = (int)((blockIdx.x * (long long)blockDim.x + threadIdx.x) >> 5);
    int lane = threadIdx.x & 31;
    if (wave >= n_edge) return;                       // uniform per wave
    int r = rel_id[wave];
    int d = dst[wave];
    const float* e = rel + (long long)r * H_DIM;
    const float* v = ent + (long long)d * H_DIM;
    float acc = 0.f;
#pragma unroll
    for (int i = 0; i < 3; ++i) {                     // 96 = 3 * 32 lanes
        int c = lane + 32 * i;
        acc += e[c] * v[c];
    }
    for (int off = 16; off > 0; off >>= 1) acc += __shfl_xor(acc, off, 32);
    if (lane == 0) {
        score[wave] = acc;
        atomicMax(seg_max_u + d, f2ord(acc));
    }
}

// K2: thread-per-edge: w = exp(score - seg_max[dst]); seg_sum[dst] += w
__global__ void k2_sum(const float* __restrict__ score, const unsigned* __restrict__ seg_max_u,
                       const int* __restrict__ dst, float* __restrict__ seg_sum, int n_edge) {
    int e = blockIdx.x * blockDim.x + threadIdx.x;
    if (e >= n_edge) return;
    int d = dst[e];
    float w = __expf(score[e] - ord2f(seg_max_u[d]));
    atomicAdd(seg_sum + d, w);
}

// K3: wave-per-edge weighted scatter: neigh[dst] += rel_emb[rel_id] * alpha
__global__ void k3_scatter(const float* __restrict__ rel, const int* __restrict__ rel_id,
                           const int* __restrict__ dst, const float* __restrict__ score,
                           const unsigned* __restrict__ seg_max_u, const float* __restrict__ seg_sum,
                           float* __restrict__ neigh, int n_edge) {
    int wave = (int)((blockIdx.x * (long long)blockDim.x + threadIdx.x) >> 5);
    int lane = threadIdx.x & 31;
    if (wave >= n_edge) return;                       // uniform per wave
    int r = rel_id[wave];
    int d = dst[wave];
    float alpha = __expf(score[wave] - ord2f(seg_max_u[d])) / seg_sum[d];
    const float* e = rel + (long long)r * H_DIM;
    float* o = neigh + (long long)d * H_DIM;
#pragma unroll
    for (int i = 0; i < 3; ++i) {
        int c = lane + 32 * i;
        atomicAdd(o + c, e[c] * alpha);
    }
}

// K4: C[0:16*m_tiles, 96] = A @ W[96,96] with V_WMMA_F32_16X16X4_F32.
// Full tiles only (no guards -> clean b64 A loads, dense WMMA clauses).
// W staged in LDS (padded stride) once per block; B frags via ds_load.
// A-frag (ISA 32-bit A 16x4 layout): lanes 0-15 -> K=0,1 ; lanes 16-31 -> K=2,3.
// C/D: VGPR i: lanes 0-15 -> M=i, lanes 16-31 -> M=i+8; N = lane&15.
__global__ void k4_gemm(const float* __restrict__ A, const float* __restrict__ W,
                        float* __restrict__ C, int m_tiles, int n_tiles) {
    __shared__ float wlds[H_DIM * WLDS_STRIDE];
    // cooperative stage of the whole 96x96 W into LDS (all threads; barrier-safe)
    for (int idx = threadIdx.x; idx < H_DIM * H_DIM; idx += blockDim.x) {
        int kk = idx / H_DIM;
        int c  = idx - kk * H_DIM;
        wlds[kk * WLDS_STRIDE + c] = W[idx];
    }
    __syncthreads();

    int wave = (int)((blockIdx.x * (long long)blockDim.x + threadIdx.x) >> 5);
    int lane = threadIdx.x & 31;
    if (wave >= m_tiles * n_tiles) return;            // uniform per wave
    int mt = wave / n_tiles;
    int nt = wave - mt * n_tiles;
    int half = lane >> 4;                             // 0 | 1
    int l    = lane & 15;
    const float* ap = A + ((long long)(mt * 16 + l)) * H_DIM + half * 2;
    int col = nt * 16 + l;

    v8f acc = {};
#pragma unroll
    for (int k = 0; k < H_DIM; k += 4) {
        int kk = k + half * 2;
        v2f a = *(const v2f*)(ap + k);                // global_load_b64, 8B aligned
        v2f b;
        b.x = wlds[kk * WLDS_STRIDE + col];
        b.y = wlds[(kk + 1) * WLDS_STRIDE + col];
        acc = __builtin_amdgcn_wmma_f32_16x16x4_f32(
            /*neg_a=*/false, a, /*neg_b=*/false, b,
            /*c_mod=*/(short)0, acc, /*reuse_a=*/false, /*reuse_b=*/false);
    }
#pragma unroll
    for (int i = 0; i < 8; ++i) {
        int row = mt * 16 + i + half * 8;
        C[(long long)row * H_DIM + col] = acc[i];
    }
}

// K4b: scalar tail for remainder rows (launches 0 blocks when n_ent % 16 == 0)
__global__ void k4_tail(const float* __restrict__ A, const float* __restrict__ W,
                        float* __restrict__ C, int row0, int n_rows) {
    int idx = blockIdx.x * blockDim.x + threadIdx.x;
    int row = row0 + idx / H_DIM;
    int col = idx % H_DIM;
    if (row >= n_rows) return;
    float s = 0.f;
    for (int k = 0; k < H_DIM; ++k) s += A[(long long)row * H_DIM + k] * W[(long long)k * H_DIM + col];
    C[(long long)row * H_DIM + col] = s;
}

// K5: per-channel sum / sum-of-squares partials. blockDim.x == 96 (thread-per-channel).
__global__ void k5_stats(const float* __restrict__ X, float* __restrict__ bsum,
                         float* __restrict__ bsq, int n_rows) {
    int c = threadIdx.x;
    int rows_per_block = (n_rows + gridDim.x - 1) / gridDim.x;
    int r0 = blockIdx.x * rows_per_block;
    int r1 = r0 + rows_per_block; if (r1 > n_rows) r1 = n_rows;
    float s = 0.f, q = 0.f;
    for (int r = r0; r < r1; ++r) {
        float x = X[(long long)r * H_DIM + c];
        s += x; q += x * x;
    }
    atomicAdd(bsum + c, s);
    atomicAdd(bsq + c, q);
}

// K6: in-place BatchNorm (biased var) + tanh.
// blockDim.x == 96: thread owns a fixed channel, computes scale/shift once,
// grid-strides over rows (no per-element modulo).
__global__ void k6_bn_tanh(float* __restrict__ X, const float* __restrict__ bsum,
                           const float* __restrict__ bsq, const float* __restrict__ gamma,
                           const float* __restrict__ beta, int n_rows) {
    int c = threadIdx.x;
    float invN  = 1.0f / (float)n_rows;
    float mu    = bsum[c] * invN;
    float var   = bsq[c] * invN - mu * mu;
    float scale = rsqrtf(var + BN_EPS) * gamma[c];
    float shift = beta[c] - mu * scale;
    for (int r = blockIdx.x; r < n_rows; r += gridDim.x) {
        long long i = (long long)r * H_DIM + c;
        X[i] = tanhf(X[i] * scale + shift);
    }
}

extern "C" void kernel_launch(void* const* d_in, const int* in_sizes, int n_in,
                              void* d_out, int out_size, void* d_ws, size_t ws_size,
                              hipStream_t stream) {
    const float* ent   = (const float*)d_in[0];
    const float* rel   = (const float*)d_in[1];
    const float* W     = (const float*)d_in[2];
    const float* gamma = (const float*)d_in[3];
    const float* beta  = (const float*)d_in[4];
    const int*  rel_id = (const int*)d_in[5];
    const int*  dst    = (const int*)d_in[6];
    int n_ent  = in_sizes[0] / H_DIM;
    int n_edge = in_sizes[5];
    float* out = (float*)d_out;

    // ---- workspace layout (contiguous zero-init prefix) ----
    char* ws = (char*)d_ws;
    unsigned* seg_max_u = (unsigned*)ws;                                     // n_ent
    float* seg_sum = (float*)(ws + (size_t)n_ent * 4);                       // n_ent
    float* neigh   = (float*)(ws + (size_t)n_ent * 8);                       // n_ent * 96
    float* bsum    = (float*)(ws + (size_t)n_ent * 8 + (size_t)n_ent * H_DIM * 4); // 96
    float* bsq     = bsum + H_DIM;                                           // 96
    float* score   = bsq + H_DIM;                                            // n_edge

    long long zero_words = 2LL * n_ent + (long long)n_ent * H_DIM + 2 * H_DIM;
    k0_zero<<<2048, 256, 0, stream>>>(seg_max_u, zero_words);

    // wave-per-edge launches: 8 waves / 256-thread block
    int edge_wave_blocks = (int)(((long long)n_edge * 32 + 255) / 256);
    k1_score<<<edge_wave_blocks, 256, 0, stream>>>(ent, rel, rel_id, dst,
                                                   score, seg_max_u, n_edge);

    k2_sum<<<(n_edge + 255) / 256, 256, 0, stream>>>(score, seg_max_u, dst,
                                                     seg_sum, n_edge);

    k3_scatter<<<edge_wave_blocks, 256, 0, stream>>>(rel, rel_id, dst, score,
                                                     seg_max_u, seg_sum, neigh, n_edge);

    // GEMM: full 16-row tiles via WMMA, remainder rows via scalar tail
    int m_tiles = n_ent / 16;               // 3125 full tiles for 50000
    int n_tiles = H_DIM / 16;               // 6
    int blocks4 = (m_tiles * n_tiles + 7) / 8;
    k4_gemm<<<blocks4, 256, 0, stream>>>(neigh, W, out, m_tiles, n_tiles);
    int row0 = m_tiles * 16;
    int tail_rows = n_ent - row0;
    if (tail_rows > 0) {
        int tail_threads = tail_rows * H_DIM;
        k4_tail<<<(tail_threads + 255) / 256, 256, 0, stream>>>(neigh, W, out, row0, n_ent);
    }

    k5_stats<<<256, H_DIM, 0, stream>>>(out, bsum, bsq, n_ent);

    k6_bn_tanh<<<2048, H_DIM, 0, stream>>>(out, bsum, bsq, gamma, beta, n_ent);
}